// Sparsify_111669149795
// MI455X (gfx1250) — compile-verified
//
#include <hip/hip_runtime.h>

// Block top-k sparsify: keep top K=4 of every BLOCK=8 scores, out = x * mask.
// Memory-bound (12 B/elem, ~541 MB stream, ~23 us floor @ 23.3 TB/s on MI455X).
// Strategy: b128 vector loads/stores with non-temporal hints (one-shot stream,
// don't pollute the 192 MB L2), global_prefetch_b8 read-ahead, all-in-lane
// 28-compare ranking network (no LDS, no cross-lane ops), wave32-sized blocks.

typedef float v4f __attribute__((ext_vector_type(4)));

#ifndef THREADS_PER_BLOCK
#define THREADS_PER_BLOCK 256   // 8 wave32 waves per workgroup
#endif
#define BLOCKS_PER_THREAD 4     // grid-stride iterations per thread

__global__ __launch_bounds__(THREADS_PER_BLOCK)
void btopk48_sparsify_kernel(const v4f* __restrict__ x4,
                             const v4f* __restrict__ s4,
                             v4f* __restrict__ o4,
                             unsigned nblk)  // number of 8-element blocks
{
    const unsigned stride = gridDim.x * blockDim.x;
    for (unsigned b = blockIdx.x * blockDim.x + threadIdx.x; b < nblk; b += stride) {
        // Prefetch next grid-stride iteration into L2 (global_prefetch_b8).
        unsigned nb = b + stride;
        if (nb < nblk) {
            __builtin_prefetch((const void*)&s4[2u * nb], 0, 0);
            __builtin_prefetch((const void*)&x4[2u * nb], 0, 0);
        }

        // Two b128 loads per array per block (8 floats), non-temporal.
        v4f slo = __builtin_nontemporal_load(&s4[2u * b]);
        v4f shi = __builtin_nontemporal_load(&s4[2u * b + 1u]);
        v4f xlo = __builtin_nontemporal_load(&x4[2u * b]);
        v4f xhi = __builtin_nontemporal_load(&x4[2u * b + 1u]);

        float s[8] = { slo[0], slo[1], slo[2], slo[3],
                       shi[0], shi[1], shi[2], shi[3] };

        // c[i] = #elements outranking i in descending order, with the
        // later-index-wins tie-break that matches stable ascending argsort:
        // for pair i<j, j outranks i iff s[j] >= s[i]. Keep iff c[i] <= 3.
        int c[8] = {0, 0, 0, 0, 0, 0, 0, 0};
#pragma unroll
        for (int i = 0; i < 8; ++i) {
#pragma unroll
            for (int j = i + 1; j < 8; ++j) {
                int t = (s[j] >= s[i]) ? 1 : 0;  // one v_cmp per pair (28 total)
                c[i] += t;
                c[j] += 1 - t;
            }
        }

        v4f olo, ohi;
#pragma unroll
        for (int k = 0; k < 4; ++k)
            olo[k] = (c[k] <= 3) ? xlo[k] : 0.0f;
#pragma unroll
        for (int k = 0; k < 4; ++k)
            ohi[k] = (c[4 + k] <= 3) ? xhi[k] : 0.0f;

        __builtin_nontemporal_store(olo, &o4[2u * b]);
        __builtin_nontemporal_store(ohi, &o4[2u * b + 1u]);
    }
}

extern "C" void kernel_launch(void* const* d_in, const int* in_sizes, int n_in,
                              void* d_out, int out_size, void* d_ws, size_t ws_size,
                              hipStream_t stream) {
    (void)n_in; (void)d_ws; (void)ws_size; (void)out_size;

    const float* x     = (const float*)d_in[0];   // fp32, SHAPE flat
    const float* score = (const float*)d_in[1];   // fp32, SHAPE flat
    float*       out   = (float*)d_out;           // fp32, SHAPE flat

    const unsigned n    = (unsigned)in_sizes[0];  // 4096*11008, multiple of 8
    const unsigned nblk = n / 8u;                 // 5,636,096 blocks

    const unsigned threads = THREADS_PER_BLOCK;
    const unsigned total_threads = (nblk + BLOCKS_PER_THREAD - 1u) / BLOCKS_PER_THREAD;
    const unsigned grid = (total_threads + threads - 1u) / threads;  // 5504

    btopk48_sparsify_kernel<<<grid, threads, 0, stream>>>(
        (const v4f*)x, (const v4f*)score, (v4f*)out, nblk);
}